// Encoder_21345987461442
// MI455X (gfx1250) — compile-verified
//
#include <hip/hip_runtime.h>

typedef float v2f __attribute__((ext_vector_type(2)));
typedef float v8f __attribute__((ext_vector_type(8)));

// ---------------- utility kernels ----------------

__global__ void zero_f32(float* __restrict__ p, long long nElems) {
  long long i = (long long)blockIdx.x * blockDim.x + threadIdx.x;
  if (i < nElems) p[i] = 0.0f;
}

__global__ void deg_accum(const int* __restrict__ dst, float* __restrict__ deg, int nE) {
  int e = blockIdx.x * blockDim.x + threadIdx.x;
  if (e < nE) atomicAdd(&deg[dst[e]], 1.0f);
}

__global__ void deg_finalize(float* __restrict__ deg, int n) {
  int i = blockIdx.x * blockDim.x + threadIdx.x;
  if (i < n) deg[i] = rsqrtf(deg[i] + 1.0f);  // dis = (deg+1)^-1/2
}

// ---------------- WMMA f32 GEMM: D = A[MxK] * B[KxN] ----------------
// One wave computes one 16x16 tile of D using V_WMMA_F32_16X16X4_F32.
// A-matrix VGPR layout (16x4 f32): lanes 0-15 -> M=lane, v0=K0,v1=K1;
// lanes 16-31 -> M=lane-16, v0=K2,v1=K3. B/C/D: rows striped across lanes.
__global__ void gemm_wmma_f32(const float* __restrict__ A, const float* __restrict__ B,
                              float* __restrict__ D, int M, int K, int N) {
  int wavesPerBlock = blockDim.x >> 5;
  long long wave = (long long)blockIdx.x * wavesPerBlock + (threadIdx.x >> 5);
  int tilesN = N >> 4;
  long long tilesM = (M + 15) >> 4;
  long long tm = wave / tilesN;
  int tn = (int)(wave % tilesN);
  if (tm >= tilesM) return;  // wave-uniform exit: EXEC stays all-1s for WMMA

  int lane = threadIdx.x & 31;
  int lr = lane & 15;   // row (A) / col (B,D) within tile
  int hi = lane >> 4;   // selects K pair 0/1 vs 2/3

  int rowA = (int)(tm * 16) + lr;
  if (rowA > M - 1) rowA = M - 1;  // clamp loads; keep all lanes active
  const float* arow = A + (size_t)rowA * K;
  const float* bcol = B + (size_t)tn * 16 + lr;

  v8f acc = {};
  for (int k = 0; k < K; k += 4) {
    v2f a, b;
    int ka = k + 2 * hi;
    a.x = arow[ka];
    a.y = arow[ka + 1];
    b.x = bcol[(size_t)ka * N];
    b.y = bcol[(size_t)(ka + 1) * N];
    acc = __builtin_amdgcn_wmma_f32_16x16x4_f32(
        /*neg_a=*/false, a, /*neg_b=*/false, b,
        /*c_mod=*/(short)0, acc, /*reuse_a=*/false, /*reuse_b=*/false);
  }

  // D layout: VGPR j -> row tm*16 + j + 8*hi, col tn*16 + lr
  float* dcol = D + (size_t)tn * 16 + lr;
#pragma unroll
  for (int j = 0; j < 8; ++j) {
    int rowD = (int)(tm * 16) + j + 8 * hi;
    if (rowD < M) dcol[(size_t)rowD * N] = acc[j];
  }
}

// ---------------- edge scatter: agg[dst] += h[src] * dis[src]*dis[dst] ----------------
// One thread per (edge, channel). Lanes within a wave share an edge (C >= 32),
// so h[src] loads are coalesced and src/dst/dis loads broadcast.
template <int C>
__global__ void scatter_add(const float* __restrict__ h, const int* __restrict__ src,
                            const int* __restrict__ dst, const float* __restrict__ dis,
                            float* __restrict__ agg, int nE) {
  long long gid = (long long)blockIdx.x * blockDim.x + threadIdx.x;
  int e = (int)(gid / C);
  if (e >= nE) return;
  int c = (int)(gid % C);
  int s = src[e];
  int d = dst[e];
  float norm = dis[s] * dis[d];
  atomicAdd(&agg[(size_t)d * C + c], h[(size_t)s * C + c] * norm);
}

// ---------------- out = agg + h * dis^2 + b, optional ReLU ----------------
template <int C, bool RELU>
__global__ void self_bias(const float* __restrict__ agg, const float* __restrict__ h,
                          const float* __restrict__ dis, const float* __restrict__ b,
                          float* __restrict__ out, int n) {
  long long gid = (long long)blockIdx.x * blockDim.x + threadIdx.x;
  if (gid >= (long long)n * C) return;
  int i = (int)(gid / C);
  int c = (int)(gid % C);
  float di = dis[i];
  float v = agg[gid] + h[gid] * (di * di) + b[c];
  out[gid] = RELU ? fmaxf(v, 0.0f) : v;
}

// ---------------- driver ----------------

extern "C" void kernel_launch(void* const* d_in, const int* in_sizes, int n_in,
                              void* d_out, int out_size, void* d_ws, size_t ws_size,
                              hipStream_t stream) {
  const float* x  = (const float*)d_in[0];
  const float* W1 = (const float*)d_in[1];
  const float* b1 = (const float*)d_in[2];
  const float* W2 = (const float*)d_in[3];
  const float* b2 = (const float*)d_in[4];
  const int*   ei = (const int*)d_in[5];

  const int IN  = 128;
  const int HID = in_sizes[2];        // 64
  const int OUT = in_sizes[4];        // 32
  const int n   = in_sizes[0] / IN;   // 50000
  const int nE  = in_sizes[5] / 2;    // 800000
  const int* src = ei;
  const int* dst = ei + nE;

  char* ws = (char*)d_ws;
  size_t off = 0;
  auto alloc = [&](size_t bytes) -> float* {
    float* p = (float*)(ws + off);
    off += (bytes + 255) & ~(size_t)255;
    return p;
  };
  float* dis  = alloc((size_t)n * 4);          // degree -> dis in place
  float* h1   = alloc((size_t)n * HID * 4);    // x@W1, then relu'd hidden
  float* agg1 = alloc((size_t)n * HID * 4);
  float* h2   = alloc((size_t)n * OUT * 4);
  float* agg2 = alloc((size_t)n * OUT * 4);
  float* outp = (float*)d_out;

  const int T = 256;  // 8 waves of 32 on CDNA5
  auto blocks1d = [&](long long work) -> unsigned int {
    return (unsigned int)((work + T - 1) / T);
  };

  // ---- degrees -> dis ----
  zero_f32<<<blocks1d(n), T, 0, stream>>>(dis, n);
  deg_accum<<<blocks1d(nE), T, 0, stream>>>(dst, dis, nE);
  deg_finalize<<<blocks1d(n), T, 0, stream>>>(dis, n);

  // ---- layer 1: h1 = x @ W1 ----
  {
    long long tilesM = (n + 15) / 16;
    long long waves = tilesM * (HID / 16);
    unsigned int blocks = (unsigned int)((waves + (T / 32) - 1) / (T / 32));
    gemm_wmma_f32<<<blocks, T, 0, stream>>>(x, W1, h1, n, IN, HID);
  }
  zero_f32<<<blocks1d((long long)n * HID), T, 0, stream>>>(agg1, (long long)n * HID);
  scatter_add<64><<<blocks1d((long long)nE * 64), T, 0, stream>>>(h1, src, dst, dis, agg1, nE);
  // h1 <- relu(agg1 + h1*dis^2 + b1) (in place over h1)
  self_bias<64, true><<<blocks1d((long long)n * 64), T, 0, stream>>>(agg1, h1, dis, b1, h1, n);

  // ---- layer 2: h2 = h1 @ W2 ----
  {
    long long tilesM = (n + 15) / 16;
    long long waves = tilesM * (OUT / 16);
    unsigned int blocks = (unsigned int)((waves + (T / 32) - 1) / (T / 32));
    gemm_wmma_f32<<<blocks, T, 0, stream>>>(h1, W2, h2, n, HID, OUT);
  }
  zero_f32<<<blocks1d((long long)n * OUT), T, 0, stream>>>(agg2, (long long)n * OUT);
  scatter_add<32><<<blocks1d((long long)nE * 32), T, 0, stream>>>(h2, src, dst, dis, agg2, nE);
  self_bias<32, false><<<blocks1d((long long)n * 32), T, 0, stream>>>(agg2, h2, dis, b2, outp, n);
}